// GraphReconstructionGCN_64527588655436
// MI455X (gfx1250) — compile-verified
//
#include <hip/hip_runtime.h>
#include <math.h>

typedef __attribute__((ext_vector_type(2))) float v2f;
typedef __attribute__((ext_vector_type(8))) float v8f;

// ---------------------------------------------------------------------------
// Norm precompute kernels
// ---------------------------------------------------------------------------
__global__ void init_deg_kernel(float* __restrict__ deg, int n) {
  int i = blockIdx.x * blockDim.x + threadIdx.x;
  if (i < n) deg[i] = 1.0f;  // self-loop weight
}

__global__ void accum_deg_kernel(const long long* __restrict__ dst,
                                 const float* __restrict__ w,
                                 float* __restrict__ deg, int e) {
  int i = blockIdx.x * blockDim.x + threadIdx.x;
  if (i < e) atomicAdd(&deg[(int)dst[i]], w[i]);
}

__global__ void dinv_kernel(float* __restrict__ deg, int n) {
  int i = blockIdx.x * blockDim.x + threadIdx.x;
  if (i < n) {
    float d = deg[i];
    deg[i] = (d > 0.0f) ? (1.0f / sqrtf(d)) : 0.0f;  // in-place: deg -> dinv
  }
}

__global__ void norm_kernel(const long long* __restrict__ src,
                            const long long* __restrict__ dst,
                            const float* __restrict__ w,
                            const float* __restrict__ dinv,
                            float* __restrict__ norm, int e) {
  int i = blockIdx.x * blockDim.x + threadIdx.x;
  if (i < e) norm[i] = dinv[(int)src[i]] * w[i] * dinv[(int)dst[i]];
}

// ---------------------------------------------------------------------------
// WMMA f32 GEMM: C[nrows x FOUT] = A[nrows x 128] * W[128 x FOUT]
// Block = FOUT/16 waves; each wave owns one 16-col tile of W for ALL of
// K=128, cached in 32 v2f register fragments (64 VGPRs), and sweeps
// RT=5 row tiles (80 rows/block; 3125 tiles = 5 * 625 blocks).
// A slab (80x128) staged in LDS with padded stride 132 (conflict-free).
// v_wmma_f32_16x16x4_f32 fragment layout (ISA 7.12.2, 32-bit 16x4 A):
//   A: lane l (m = l&15, half = l>>4) holds {A[m][k0+2h], A[m][k0+2h+1]}
//   B: lane l (n = l&15, half = l>>4) holds {B[k0+2h][n], B[k0+2h+1][n]}
//   C: vgpr r, lane l -> C[r + 8*(l>>4)][l&15]
// ---------------------------------------------------------------------------
#define RT 5  // row tiles per block

template <int FOUT>
__launch_bounds__(FOUT * 2)
__global__ void gemm_wmma_kernel(const float* __restrict__ A,
                                 const float* __restrict__ W,
                                 float* __restrict__ C) {
  __shared__ float As[RT * 16 * 132];
  const int tid  = threadIdx.x;
  const int wave = tid >> 5;
  const int lane = tid & 31;
  const int row0 = blockIdx.x * (RT * 16);

  // Stage 80x128 slab of A, coalesced, padded row stride 132
  for (int i = tid; i < RT * 16 * 128; i += FOUT * 2) {
    int r = i >> 7, c = i & 127;
    As[r * 132 + c] = A[(size_t)row0 * 128 + i];
  }

  const int n0   = wave * 16;
  const int mn   = lane & 15;   // m for A-frag, n for B-frag / C store
  const int half = lane >> 4;

  // Cache the wave's full B column-tile in registers: 32 k-steps x v2f
  v2f bfrag[32];
#pragma unroll
  for (int s = 0; s < 32; ++s) {
    const int ka = 4 * s + 2 * half;
    bfrag[s].x = W[(size_t)(ka + 0) * FOUT + n0 + mn];
    bfrag[s].y = W[(size_t)(ka + 1) * FOUT + n0 + mn];
  }

  __syncthreads();

  for (int rt = 0; rt < RT; ++rt) {
    const float* as = &As[(rt * 16 + mn) * 132 + 2 * half];
    v8f acc = {};
#pragma unroll
    for (int s = 0; s < 32; ++s) {
      v2f a;
      a.x = as[4 * s + 0];
      a.y = as[4 * s + 1];
      acc = __builtin_amdgcn_wmma_f32_16x16x4_f32(
          /*neg_a=*/false, a, /*neg_b=*/false, bfrag[s],
          /*c_mod=*/(short)0, acc, /*reuse_a=*/false, /*reuse_b=*/false);
    }
    float* cp = C + (size_t)(row0 + rt * 16) * FOUT + n0;
#pragma unroll
    for (int r = 0; r < 8; ++r) {
      int m = r + half * 8;
      cp[(size_t)m * FOUT + mn] = acc[r];
    }
  }
}

// ---------------------------------------------------------------------------
// Self-loop init + bias: out[i,f] = dinv[i]^2 * h[i,f] + b[f]
// ---------------------------------------------------------------------------
template <int F>
__global__ void selfloop_bias_kernel(const float* __restrict__ h,
                                     const float* __restrict__ dinv,
                                     const float* __restrict__ b,
                                     float* __restrict__ out, int total) {
  int idx = blockIdx.x * blockDim.x + threadIdx.x;
  if (idx < total) {
    int i = idx / F, f = idx % F;
    float di = dinv[i];
    out[idx] = di * di * h[idx] + b[f];
  }
}

// ---------------------------------------------------------------------------
// Edge scatter: 256-thread blocks, each handling 256/F edges; lanes cover
// features (fully coalesced); atomicAdd into L2-resident accumulator.
// ---------------------------------------------------------------------------
template <int F>
__launch_bounds__(256)
__global__ void scatter_edges_kernel(const long long* __restrict__ src,
                                     const long long* __restrict__ dst,
                                     const float* __restrict__ norm,
                                     const float* __restrict__ h,
                                     float* __restrict__ out, int E) {
  const int epb = 256 / F;
  const int e = blockIdx.x * epb + threadIdx.x / F;
  const int f = threadIdx.x % F;
  if (e < E) {
    const float w = norm[e];
    const int s = (int)src[e];
    const int d = (int)dst[e];
    atomicAdd(&out[(size_t)d * F + f], w * h[(size_t)s * F + f]);
  }
}

__global__ void relu_kernel(float* __restrict__ x, int total) {
  int i = blockIdx.x * blockDim.x + threadIdx.x;
  if (i < total) x[i] = fmaxf(x[i], 0.0f);
}

// ---------------------------------------------------------------------------
// Launcher
// ---------------------------------------------------------------------------
extern "C" void kernel_launch(void* const* d_in, const int* in_sizes, int n_in,
                              void* d_out, int out_size, void* d_ws, size_t ws_size,
                              hipStream_t stream) {
  const float*     x   = (const float*)d_in[0];
  const long long* ei  = (const long long*)d_in[1];  // int64 [2, E]
  const float*     ew  = (const float*)d_in[2];
  const float*     W1  = (const float*)d_in[3];
  const float*     b1  = (const float*)d_in[4];
  const float*     W2  = (const float*)d_in[5];
  const float*     b2  = (const float*)d_in[6];
  float*           out = (float*)d_out;

  const int N = in_sizes[0] / 128;   // 50000
  const int E = in_sizes[2];         // 800000
  const long long* src = ei;         // row 0
  const long long* dst = ei + E;     // row 1

  // Workspace layout (floats), 256-element aligned blocks
  float* ws = (float*)d_ws;
  size_t NP = ((size_t)N + 255) & ~(size_t)255;
  size_t EP = ((size_t)E + 255) & ~(size_t)255;
  float* dinv = ws;                       // N
  float* norm = dinv + NP;                // E
  float* h1   = norm + EP;                // N*128
  float* agg1 = h1 + (size_t)N * 128;     // N*128
  float* h2   = h1;                       // alias: h1 dead once agg1 is final

  const int TPB = 256;

  // ---- gcn_norm ----
  init_deg_kernel<<<(N + TPB - 1) / TPB, TPB, 0, stream>>>(dinv, N);
  accum_deg_kernel<<<(E + TPB - 1) / TPB, TPB, 0, stream>>>(dst, ew, dinv, E);
  dinv_kernel<<<(N + TPB - 1) / TPB, TPB, 0, stream>>>(dinv, N);
  norm_kernel<<<(E + TPB - 1) / TPB, TPB, 0, stream>>>(src, dst, ew, dinv, norm, E);

  // ---- layer 1: h1 = x @ W1 ; agg1 = scatter(norm * h1) + b1 ; relu ----
  gemm_wmma_kernel<128><<<N / (16 * RT), 256, 0, stream>>>(x, W1, h1);
  selfloop_bias_kernel<128><<<(N * 128 + TPB - 1) / TPB, TPB, 0, stream>>>(
      h1, dinv, b1, agg1, N * 128);
  scatter_edges_kernel<128><<<(E + 1) / 2, 256, 0, stream>>>(src, dst, norm, h1, agg1, E);
  relu_kernel<<<(N * 128 + TPB - 1) / TPB, TPB, 0, stream>>>(agg1, N * 128);

  // ---- layer 2: h2 = agg1 @ W2 ; out = scatter(norm * h2) + b2 ----
  gemm_wmma_kernel<64><<<N / (16 * RT), 128, 0, stream>>>(agg1, W2, h2);
  selfloop_bias_kernel<64><<<(N * 64 + TPB - 1) / TPB, TPB, 0, stream>>>(
      h2, dinv, b2, out, N * 64);
  scatter_edges_kernel<64><<<(E + 3) / 4, 256, 0, stream>>>(src, dst, norm, h2, out, E);
}